// RGCN_22187801051464
// MI455X (gfx1250) — compile-verified
//
#include <hip/hip_runtime.h>
#include <cstdint>
#include <cstddef>

typedef float v2f __attribute__((ext_vector_type(2)));
typedef float v8f __attribute__((ext_vector_type(8)));
typedef uint32_t v4u __attribute__((ext_vector_type(4)));
typedef int v4i __attribute__((ext_vector_type(4)));
typedef int v8i __attribute__((ext_vector_type(8)));

#define ASTR 132  // padded LDS stride for the 64x128 A tile (bank-conflict-free)

__device__ __forceinline__ void atomic_add_f32(float* p, float v) {
  unsafeAtomicAdd(p, v);  // lowers to global_atomic_add_f32 on gfx1250
}

// CDNA5 async memory->LDS copy, 16 bytes per active lane (ASYNCcnt-tracked).
__device__ __forceinline__ void async_copy_b128(uint32_t lds_addr, const void* gaddr) {
  asm volatile("global_load_async_to_lds_b128 %0, %1, off"
               :: "v"(lds_addr), "v"(gaddr)
               : "memory");
}
__device__ __forceinline__ void async_wait0() {
  asm volatile("s_wait_asynccnt 0x0" ::: "memory");
}
__device__ __forceinline__ uint32_t lds_u32(const void* p) {
  return (uint32_t)(uintptr_t)p;  // flat LDS addr truncates to LDS byte offset
}

// Tensor Data Mover: 2D tile (rows x dim0 f32) global->LDS with per-row LDS padding.
// Issue from ONE wave only (TDM ignores EXEC; every executing wave issues a DMA).
// pad_icode: pad fires every 2<<pad_icode dwords; pad_acode: inserts pad_acode+1 dwords.
__device__ __forceinline__ void tdm_load_2d(uint32_t lds_addr, const void* gaddr,
                                            uint32_t dim0, uint32_t tensor_rows,
                                            uint32_t tile_rows, uint32_t pad_icode,
                                            uint32_t pad_acode) {
  uint64_t ga = (uint64_t)(uintptr_t)gaddr;
  v4u g0;
  g0[0] = 1u;                                               // count=1, user mode
  g0[1] = lds_addr;                                         // lds_addr
  g0[2] = (uint32_t)ga;                                     // global_addr[31:0]
  g0[3] = (uint32_t)((ga >> 32) & 0x01FFFFFFu) | (2u << 30);  // addr[56:32] | type=2
  v8i g1;
  g1[0] = (int)((2u << 16) | (1u << 20) | (pad_icode << 22) | (pad_acode << 25));
  g1[1] = (int)((dim0 & 0xFFFFu) << 16);                    // tensor_dim0[15:0]
  g1[2] = (int)(((dim0 >> 16) & 0xFFFFu) | ((tensor_rows & 0xFFFFu) << 16));
  g1[3] = (int)(((tensor_rows >> 16) & 0xFFFFu) | ((dim0 & 0xFFFFu) << 16));  // tile_dim0
  g1[4] = (int)(tile_rows & 0xFFFFu);                       // tile_dim1 (tile_dim2=0)
  g1[5] = (int)dim0;                                        // tensor_dim0_stride[31:0]
  g1[6] = (int)((dim0 >> 16) & 0xFFFFu);                    // stride[47:32] | dim1_stride lo=0
  g1[7] = 0;
  v4i gz = {0, 0, 0, 0};
#if __clang_major__ >= 23
  v8i gz8 = {0, 0, 0, 0, 0, 0, 0, 0};
  __builtin_amdgcn_tensor_load_to_lds(g0, g1, gz, gz, gz8, 0);
#else
  __builtin_amdgcn_tensor_load_to_lds(g0, g1, gz, gz, 0);
#endif
}
__device__ __forceinline__ void tdm_wait0() {
  __builtin_amdgcn_s_wait_tensorcnt(0);
}

// ---------------- W[r] = sum_b a[r,b] * V[b]  ----------------
__global__ void combine_bases_kernel(const float* __restrict__ V,
                                     const float* __restrict__ a,
                                     float* __restrict__ W,
                                     int per_rel, int total) {
  int i = blockIdx.x * blockDim.x + threadIdx.x;
  if (i >= total) return;
  int r = i / per_rel;
  int j = i - r * per_rel;
  float s = 0.f;
#pragma unroll
  for (int b = 0; b < 8; ++b) s += a[r * 8 + b] * V[(size_t)b * per_rel + j];
  W[i] = s;
}

// ---------------- edge bucketing by relation ----------------
__global__ void hist_kernel(const int* __restrict__ etype, int E, int* __restrict__ hist) {
  int e = blockIdx.x * blockDim.x + threadIdx.x;
  if (e < E) atomicAdd(&hist[etype[e]], 1);
}

__global__ void scan_kernel(const int* __restrict__ hist, int* __restrict__ off,
                            int* __restrict__ cursor) {
  if (threadIdx.x == 0 && blockIdx.x == 0) {
    int acc = 0;
    off[0] = 0;
    for (int r = 0; r < 64; ++r) {
      cursor[r] = acc;
      acc += (hist[r] + 63) & ~63;  // pad each relation segment to a 64-edge tile
      off[r + 1] = acc;
    }
  }
}

__global__ void scatter_kernel(const int* __restrict__ etype, int E,
                               int* __restrict__ cursor, int* __restrict__ perm) {
  int e = blockIdx.x * blockDim.x + threadIdx.x;
  if (e < E) {
    int p = atomicAdd(&cursor[etype[e]], 1);
    perm[p] = e;
  }
}

// ---------------- edge-tile GEMM: 64 edges x (128 -> OUT), one relation ----------------
// block = 2*OUT threads = OUT/16 waves; wave w owns output columns [16w,16w+16)
template <int OUT>
__global__ void edge_gemm_kernel(const float* __restrict__ H,    // [N,128]
                                 const float* __restrict__ W,    // [64,128,OUT]
                                 const int* __restrict__ perm,   // padded, -1 = invalid
                                 const int* __restrict__ poff,   // [65] padded offsets
                                 const int* __restrict__ hist,   // [64] true counts
                                 const int* __restrict__ src,
                                 const int* __restrict__ dst,
                                 const float* __restrict__ norm,
                                 const float* __restrict__ zpage,  // 256B of zeros
                                 float* __restrict__ out) {        // [N,OUT] atomic acc
  constexpr int WSTR = OUT + 16;
  extern __shared__ float smem[];
  float* Alds = smem;                          // 64 x ASTR
  float* Wlds = smem + 64 * ASTR;              // 128 x WSTR
  int* sS = (int*)(Wlds + 128 * WSTR);         // 64
  int* dS = sS + 64;                           // 64
  float* nS = (float*)(dS + 64);               // 64

  const int tid = threadIdx.x;
  const int start = blockIdx.x * 64;
  if (start >= poff[64]) return;
  int rel = 63;
  for (int i = 0; i < 64; ++i)
    if (start < poff[i + 1]) { rel = i; break; }
  const bool full = (start + 64) <= (poff[rel] + hist[rel]);  // whole tile valid?

  // kick off W[rel] (128 x OUT -> padded-stride LDS) on the Tensor Data Mover
  if (tid < 32) {
    tdm_load_2d(lds_u32(Wlds), W + (size_t)rel * 128 * OUT, OUT, 128, 128,
                (OUT == 128) ? 6u : 5u, 15u);  // pad 16 dwords after every OUT dwords
  }

  for (int i = tid; i < 64; i += blockDim.x) {
    int e = perm[start + i];
    int s = -1, d = 0;
    float nm = 0.f;
    if (e >= 0) { s = src[e]; d = dst[e]; nm = norm[e]; }
    sS[i] = s; dS[i] = d; nS[i] = nm;
  }
  __syncthreads();

  // async-gather A: 64 edge rows x 128 feats (invalid rows pull the zero page)
  for (int idx = tid; idx < 64 * 32; idx += blockDim.x) {
    int row = idx >> 5;
    int col = (idx & 31) * 4;
    int s = sS[row];
    const float* gp = (s >= 0) ? (H + (size_t)s * 128 + col) : zpage;
    async_copy_b128(lds_u32(Alds + row * ASTR + col), gp);
  }
  async_wait0();
  if (tid < 32) tdm_wait0();
  __syncthreads();

  const int lane = tid & 31;
  const int wave = tid >> 5;
  const int nb = wave * 16;
  const int m = lane & 15;
  const int kb = (lane < 16) ? 0 : 2;  // K split across lane halves (A/B frag layout)

  v8f c0 = {}, c1 = {}, c2 = {}, c3 = {};
#pragma unroll 4
  for (int k = 0; k < 128; k += 4) {
    v2f b;
    b.x = Wlds[(k + kb) * WSTR + nb + m];
    b.y = Wlds[(k + kb + 1) * WSTR + nb + m];
    v2f fa0, fa1, fa2, fa3;
    fa0.x = Alds[(0 + m) * ASTR + k + kb];  fa0.y = Alds[(0 + m) * ASTR + k + kb + 1];
    fa1.x = Alds[(16 + m) * ASTR + k + kb]; fa1.y = Alds[(16 + m) * ASTR + k + kb + 1];
    fa2.x = Alds[(32 + m) * ASTR + k + kb]; fa2.y = Alds[(32 + m) * ASTR + k + kb + 1];
    fa3.x = Alds[(48 + m) * ASTR + k + kb]; fa3.y = Alds[(48 + m) * ASTR + k + kb + 1];
    c0 = __builtin_amdgcn_wmma_f32_16x16x4_f32(false, fa0, false, b, (short)0, c0, false, false);
    c1 = __builtin_amdgcn_wmma_f32_16x16x4_f32(false, fa1, false, b, (short)0, c1, false, false);
    c2 = __builtin_amdgcn_wmma_f32_16x16x4_f32(false, fa2, false, b, (short)0, c2, false, false);
    c3 = __builtin_amdgcn_wmma_f32_16x16x4_f32(false, fa3, false, b, (short)0, c3, false, false);
  }

  const int n = nb + m;
  const int hrow = (lane < 16) ? 0 : 8;  // C/D layout: vgpr i -> row i (+8 for hi lanes)
  if (full) {
#pragma unroll
    for (int i = 0; i < 8; ++i) {
      int er;
      er = 0 + i + hrow;
      atomic_add_f32(out + (size_t)dS[er] * OUT + n, c0[i] * nS[er]);
      er = 16 + i + hrow;
      atomic_add_f32(out + (size_t)dS[er] * OUT + n, c1[i] * nS[er]);
      er = 32 + i + hrow;
      atomic_add_f32(out + (size_t)dS[er] * OUT + n, c2[i] * nS[er]);
      er = 48 + i + hrow;
      atomic_add_f32(out + (size_t)dS[er] * OUT + n, c3[i] * nS[er]);
    }
  } else {
#pragma unroll
    for (int i = 0; i < 8; ++i) {
      int er;
      er = 0 + i + hrow;
      if (sS[er] >= 0) atomic_add_f32(out + (size_t)dS[er] * OUT + n, c0[i] * nS[er]);
      er = 16 + i + hrow;
      if (sS[er] >= 0) atomic_add_f32(out + (size_t)dS[er] * OUT + n, c1[i] * nS[er]);
      er = 32 + i + hrow;
      if (sS[er] >= 0) atomic_add_f32(out + (size_t)dS[er] * OUT + n, c2[i] * nS[er]);
      er = 48 + i + hrow;
      if (sS[er] >= 0) atomic_add_f32(out + (size_t)dS[er] * OUT + n, c3[i] * nS[er]);
    }
  }
}

// ---------------- self-loop GEMM: acc[n] = (relu?)(acc[n] + X[n]@Wl + bias) ----------------
template <int OUT, bool RELU>
__global__ void node_gemm_kernel(const float* __restrict__ X,     // [N,128]
                                 const float* __restrict__ Wl,    // [128,OUT]
                                 const float* __restrict__ bias,  // [OUT]
                                 float* __restrict__ acc,         // [N,OUT] in/out
                                 int N) {
  constexpr int WSTR = OUT + 16;
  extern __shared__ float smem[];
  float* Alds = smem;
  float* Wlds = smem + 64 * ASTR;

  const int tid = threadIdx.x;
  const int base = blockIdx.x * 64;

  // both tiles via TDM from wave 0; rows past N read as zero (tensor OOB semantics)
  if (tid < 32) {
    tdm_load_2d(lds_u32(Alds), X + (size_t)base * 128, 128, (uint32_t)(N - base), 64,
                6u, 3u);  // pad 4 dwords after every 128 dwords (ASTR=132)
    tdm_load_2d(lds_u32(Wlds), Wl, OUT, 128, 128,
                (OUT == 128) ? 6u : 5u, 15u);  // pad 16 dwords after every OUT dwords
    tdm_wait0();
  }
  __syncthreads();

  const int lane = tid & 31;
  const int wave = tid >> 5;
  const int nb = wave * 16;
  const int m = lane & 15;
  const int kb = (lane < 16) ? 0 : 2;

  v8f c0 = {}, c1 = {}, c2 = {}, c3 = {};
#pragma unroll 4
  for (int k = 0; k < 128; k += 4) {
    v2f b;
    b.x = Wlds[(k + kb) * WSTR + nb + m];
    b.y = Wlds[(k + kb + 1) * WSTR + nb + m];
    v2f fa0, fa1, fa2, fa3;
    fa0.x = Alds[(0 + m) * ASTR + k + kb];  fa0.y = Alds[(0 + m) * ASTR + k + kb + 1];
    fa1.x = Alds[(16 + m) * ASTR + k + kb]; fa1.y = Alds[(16 + m) * ASTR + k + kb + 1];
    fa2.x = Alds[(32 + m) * ASTR + k + kb]; fa2.y = Alds[(32 + m) * ASTR + k + kb + 1];
    fa3.x = Alds[(48 + m) * ASTR + k + kb]; fa3.y = Alds[(48 + m) * ASTR + k + kb + 1];
    c0 = __builtin_amdgcn_wmma_f32_16x16x4_f32(false, fa0, false, b, (short)0, c0, false, false);
    c1 = __builtin_amdgcn_wmma_f32_16x16x4_f32(false, fa1, false, b, (short)0, c1, false, false);
    c2 = __builtin_amdgcn_wmma_f32_16x16x4_f32(false, fa2, false, b, (short)0, c2, false, false);
    c3 = __builtin_amdgcn_wmma_f32_16x16x4_f32(false, fa3, false, b, (short)0, c3, false, false);
  }

  const int n = nb + m;
  const int hrow = (lane < 16) ? 0 : 8;
  const float bn = bias[n];
#pragma unroll
  for (int i = 0; i < 8; ++i) {
    float cv[4] = {c0[i], c1[i], c2[i], c3[i]};
#pragma unroll
    for (int t = 0; t < 4; ++t) {
      int row = base + t * 16 + i + hrow;
      if (row < N) {
        size_t o = (size_t)row * OUT + n;
        float v = acc[o] + cv[t] + bn;
        if (RELU) v = fmaxf(v, 0.f);
        acc[o] = v;
      }
    }
  }
}

// ---------------- host launcher ----------------
extern "C" void kernel_launch(void* const* d_in, const int* in_sizes, int n_in,
                              void* d_out, int out_size, void* d_ws, size_t ws_size,
                              hipStream_t stream) {
  const float* x = (const float*)d_in[0];
  const int* src = (const int*)d_in[1];
  const int* dst = (const int*)d_in[2];
  const int* etype = (const int*)d_in[3];
  const float* norm = (const float*)d_in[4];
  const float* V1 = (const float*)d_in[5];
  const float* a1 = (const float*)d_in[6];
  const float* loop1 = (const float*)d_in[7];
  const float* b1 = (const float*)d_in[8];
  const float* V2 = (const float*)d_in[9];
  const float* a2 = (const float*)d_in[10];
  const float* loop2 = (const float*)d_in[11];
  const float* b2 = (const float*)d_in[12];
  float* out = (float*)d_out;

  const int N = in_sizes[0] / 128;
  const int E = in_sizes[1];
  const int EP = E + 64 * 64;  // capacity for per-relation padding to 64-edge tiles

  // carve workspace
  char* base = (char*)d_ws;
  size_t o = 0;
  auto carve = [&](size_t bytes) -> char* {
    char* p = base + o;
    o += (bytes + 255) & ~(size_t)255;
    return p;
  };
  float* W1f = (float*)carve((size_t)64 * 128 * 128 * 4);
  float* W2f = (float*)carve((size_t)64 * 128 * 64 * 4);
  float* h1 = (float*)carve((size_t)N * 128 * 4);
  int* perm = (int*)carve((size_t)EP * 4);
  int* meta = (int*)carve(193 * 4);
  float* zpage = (float*)carve(256);
  int* hist = meta;
  int* poff = meta + 64;     // 65 entries
  int* cursor = meta + 129;  // 64 entries

  hipMemsetAsync(h1, 0, (size_t)N * 128 * 4, stream);
  hipMemsetAsync(out, 0, (size_t)out_size * 4, stream);
  hipMemsetAsync(meta, 0, 193 * 4, stream);
  hipMemsetAsync(zpage, 0, 256, stream);
  hipMemsetAsync(perm, 0xFF, (size_t)EP * 4, stream);  // -1 sentinels

  // combine bases into per-relation weights
  {
    int t1 = 64 * 128 * 128;
    combine_bases_kernel<<<(t1 + 255) / 256, 256, 0, stream>>>(V1, a1, W1f, 128 * 128, t1);
    int t2 = 64 * 128 * 64;
    combine_bases_kernel<<<(t2 + 255) / 256, 256, 0, stream>>>(V2, a2, W2f, 128 * 64, t2);
  }

  // bucket edges by relation (padded counting sort)
  hist_kernel<<<(E + 255) / 256, 256, 0, stream>>>(etype, E, hist);
  scan_kernel<<<1, 64, 0, stream>>>(hist, poff, cursor);
  scatter_kernel<<<(E + 255) / 256, 256, 0, stream>>>(etype, E, cursor, perm);

  const int tiles = (EP + 63) / 64;
  const size_t smem_e128 = (size_t)(64 * ASTR + 128 * (128 + 16)) * 4 + 3 * 64 * 4;
  const size_t smem_e64 = (size_t)(64 * ASTR + 128 * (64 + 16)) * 4 + 3 * 64 * 4;
  const size_t smem_n128 = (size_t)(64 * ASTR + 128 * (128 + 16)) * 4;
  const size_t smem_n64 = (size_t)(64 * ASTR + 128 * (64 + 16)) * 4;

  // layer 1: edge messages (128->128) + self-loop + bias + relu
  edge_gemm_kernel<128><<<tiles, 256, smem_e128, stream>>>(x, W1f, perm, poff, hist, src, dst,
                                                           norm, zpage, h1);
  node_gemm_kernel<128, true><<<(N + 63) / 64, 256, smem_n128, stream>>>(x, loop1, b1, h1, N);

  // layer 2: edge messages (128->64) + self-loop + bias
  edge_gemm_kernel<64><<<tiles, 128, smem_e64, stream>>>(h1, W2f, perm, poff, hist, src, dst,
                                                         norm, zpage, out);
  node_gemm_kernel<64, false><<<(N + 63) / 64, 128, smem_n64, stream>>>(h1, loop2, b2, out, N);
}